// GSF_28690381538104
// MI455X (gfx1250) — compile-verified
//
#include <hip/hip_runtime.h>
#include <hip/hip_bf16.h>

// ---------------------------------------------------------------------------
// GSF forward for MI455X (gfx1250, wave32).
// fp32 end-to-end (workload is HBM-bandwidth bound; 2.8 GFLOP of conv work
// mapped onto V_WMMA_F32_16X16X4_F32 via a tap-GEMM decomposition).
// CDNA5 paths used: v_wmma_f32_16x16x4_f32 (matrix pipe),
// global_load_async_to_lds_b128 + s_wait_asynccnt (async LDS staging).
// ---------------------------------------------------------------------------

typedef __attribute__((ext_vector_type(2))) float v2f;
typedef __attribute__((ext_vector_type(8))) float v8f;

#define T_SEG   8
#define B_BATCH 8
#define C_TOT   256
#define C_HALF  128
#define H_DIM   56
#define W_DIM   56
#define HW      (H_DIM * W_DIM)          // 3136
#define HW4     (HW / 4)                 // 784
#define KTAPS   27                        // 3x3x3
#define KPAD    32                        // padded tap rows for WMMA M tiles
#define INV_HW  (1.0f / 3136.0f)

// ---------------------------------------------------------------------------
// Kernel 0: precompute BN scale/shift and the transposed, tap-padded weight
// matrix Wt[g][k(32)][ci(128)]  (k >= 27 rows are zero).
// w3d layout: (2, 128, 3,3,3) -> w3d[(g*128+ci)*27 + k]
// ---------------------------------------------------------------------------
__global__ void gsf_prep(const float* __restrict__ gamma,
                         const float* __restrict__ beta,
                         const float* __restrict__ mean,
                         const float* __restrict__ var,
                         const float* __restrict__ w3d,
                         float* __restrict__ Wt,      // (2,32,128)
                         float* __restrict__ scale,   // (256)
                         float* __restrict__ shift)   // (256)
{
    int i = blockIdx.x * 256 + threadIdx.x;
    if (i < C_TOT) {
        float s = gamma[i] * rsqrtf(var[i] + 1e-5f);
        scale[i] = s;
        shift[i] = beta[i] - mean[i] * s;
    }
    if (i < 2 * KPAD * C_HALF) {
        int ci = i & 127;
        int k  = (i >> 7) & 31;
        int g  = i >> 12;
        Wt[i] = (k < KTAPS) ? w3d[(g * C_HALF + ci) * KTAPS + k] : 0.0f;
    }
}

// ---------------------------------------------------------------------------
// Kernel 1: tap-GEMM with WMMA f32 16x16x4.
//   P[b][g][k][t][h][w] = sum_ci Wt[g][k][ci] * relu(bn(x[b, g*128+ci, t,h,w]))
// Block = 128 threads (4 waves); block -> one (b,g,t,h) row; wave -> 16 w cols.
// Weights staged in LDS via async DMA (global_load_async_to_lds_b128);
// B operand (4ch x 16 pts) is the BN+ReLU'd input loaded straight from
// global (fully coalesced over w).
//
// Assumed fp32 fragment layouts (per CDNA5 ISA §7.12.2):
//   A 16x4 : v0 = A[m=lane%16][K=(lane/16)*2],   v1 = A[m][K=(lane/16)*2+1]
//   B  4x16: v0 = B[K=(lane/16)*2][n=lane%16],   v1 = B[K=(lane/16)*2+1][n]
//   D 16x16: vgpr i -> row (lane<16 ? i : 8+i), col = lane%16
// ---------------------------------------------------------------------------
__global__ void gsf_gate_gemm(const float* __restrict__ x,
                              const float* __restrict__ Wt,
                              const float* __restrict__ scale,
                              const float* __restrict__ shift,
                              float* __restrict__ P)
{
    __shared__ __align__(16) float lA[KPAD * C_HALF];   // 16 KB

    const int h  = blockIdx.x;                    // 0..55
    const int t  = blockIdx.y;                    // 0..7
    const int bg = blockIdx.z;                    // 0..15
    const int b  = bg >> 1;
    const int g  = bg & 1;

    // stage this group's 32x128 weight matrix into LDS with async DMA:
    // 128 lanes x 16B x 8 issues = 16 KB, tracked on ASYNCcnt.
    {
        const float* src = Wt + g * KPAD * C_HALF;
        #pragma unroll
        for (int i = threadIdx.x * 4; i < KPAD * C_HALF; i += 128 * 4) {
            unsigned     lds_off = (unsigned)(unsigned long long)(&lA[i]);
            const float* gptr    = src + i;
            asm volatile("global_load_async_to_lds_b128 %0, %1, off"
                         :
                         : "v"(lds_off), "v"(gptr)
                         : "memory");
        }
        asm volatile("s_wait_asynccnt 0" ::: "memory");
    }
    __syncthreads();

    const int wave = threadIdx.x >> 5;            // 0..3 -> w-chunk
    const int lane = threadIdx.x & 31;
    const int n    = lane & 15;                   // column in tile
    const int kk   = lane >> 4;                   // 0/1 half of K/M striping
    const int w0   = wave * 16;
    const int wl   = (w0 + n < W_DIM) ? (w0 + n) : (W_DIM - 1);   // clamped load col

    const long xbase = (((long)(b * T_SEG + t) * C_TOT + g * C_HALF) * (long)HW)
                       + (long)h * W_DIM + wl;

    v8f d0 = {0.f,0.f,0.f,0.f,0.f,0.f,0.f,0.f};
    v8f d1 = {0.f,0.f,0.f,0.f,0.f,0.f,0.f,0.f};

    #pragma unroll 4
    for (int c0 = 0; c0 < C_HALF; c0 += 4) {
        const int ch = c0 + kk * 2;               // this lane's two channels
        const int gc = g * C_HALF + ch;

        // B fragment: BN+ReLU fused on load
        float b0 = x[xbase + (long)ch * HW];
        float b1 = x[xbase + (long)(ch + 1) * HW];
        b0 = fmaxf(b0 * scale[gc]     + shift[gc],     0.0f);
        b1 = fmaxf(b1 * scale[gc + 1] + shift[gc + 1], 0.0f);
        v2f bf = {b0, b1};

        // A fragments for the two 16-row tap tiles
        v2f a0 = { lA[n * C_HALF + ch],        lA[n * C_HALF + ch + 1] };
        v2f a1 = { lA[(16 + n) * C_HALF + ch], lA[(16 + n) * C_HALF + ch + 1] };

        d0 = __builtin_amdgcn_wmma_f32_16x16x4_f32(false, a0, false, bf,
                                                   (short)0, d0, false, false);
        d1 = __builtin_amdgcn_wmma_f32_16x16x4_f32(false, a1, false, bf,
                                                   (short)0, d1, false, false);
    }

    // store tap planes: P[((b*2+g)*32 + k)*T*H*W + ...]
    const int wcol = w0 + n;
    if (wcol < W_DIM) {
        const long pbase = ((((long)bg * KPAD) * T_SEG + t) * H_DIM + h) * W_DIM + wcol;
        const long kstr  = (long)T_SEG * HW;
        const int  rbase = kk * 8;                // lanes >=16 hold rows 8..15
        #pragma unroll
        for (int i = 0; i < 8; ++i) {
            int k0 = rbase + i;
            if (k0 < KTAPS)      P[pbase + (long)k0 * kstr]        = d0[i];
            int k1 = 16 + rbase + i;
            if (k1 < KTAPS)      P[pbase + (long)k1 * kstr]        = d1[i];
        }
    }
}

// ---------------------------------------------------------------------------
// Kernel 2: gather the 27 shifted tap planes -> gate = tanh(conv + bias)
// gate layout: (b, g, t, h, w)
// ---------------------------------------------------------------------------
__global__ void gsf_gate_reduce(const float* __restrict__ P,
                                const float* __restrict__ b3d,
                                float* __restrict__ gate)
{
    int idx = blockIdx.x * 256 + threadIdx.x;
    const int total = B_BATCH * 2 * T_SEG * HW;
    if (idx >= total) return;

    int w = idx % W_DIM;  int tmp = idx / W_DIM;
    int h = tmp % H_DIM;  tmp /= H_DIM;
    int t = tmp % T_SEG;  tmp /= T_SEG;
    int g = tmp & 1;
    int b = tmp >> 1;

    const long gstr = (long)KPAD * T_SEG * HW;
    const float* Pg = P + (long)(b * 2 + g) * gstr;

    float acc = b3d[g];
    #pragma unroll
    for (int kd = 0; kd < 3; ++kd) {
        int ts = t + kd - 1;
        if (ts < 0 || ts >= T_SEG) continue;
        #pragma unroll
        for (int kh = 0; kh < 3; ++kh) {
            int hs = h + kh - 1;
            if (hs < 0 || hs >= H_DIM) continue;
            #pragma unroll
            for (int kw = 0; kw < 3; ++kw) {
                int ws = w + kw - 1;
                if (ws < 0 || ws >= W_DIM) continue;
                int k = kd * 9 + kh * 3 + kw;
                acc += Pg[(((long)k * T_SEG + ts) * H_DIM + hs) * W_DIM + ws];
            }
        }
    }
    gate[idx] = tanhf(acc);
}

// ---------------------------------------------------------------------------
// Kernel 3: per-(b,c,t) spatial sums  Sx = sum x,  Sy = sum gate*x.
// float4 (b128) streaming; one block of 256 threads per (b,c,t).
// ---------------------------------------------------------------------------
__global__ void gsf_sums(const float* __restrict__ x,
                         const float* __restrict__ gate,
                         float* __restrict__ Sy,
                         float* __restrict__ Sx)
{
    __shared__ float sdy[256];
    __shared__ float sdx[256];

    const int id = blockIdx.x;                    // (b*256 + c)*8 + t
    const int t  = id & 7;
    const int c  = (id >> 3) & 255;
    const int b  = id >> 11;
    const int g  = c >> 7;

    const float* xp = x    + ((long)(b * T_SEG + t) * C_TOT + c) * (long)HW;
    const float* gp = gate + ((long)(b * 2 + g) * T_SEG + t) * (long)HW;

    float sy = 0.f, sx = 0.f;
    for (int i = threadIdx.x * 4; i < HW; i += 256 * 4) {
        float4 xv = *(const float4*)(xp + i);
        float4 gv = *(const float4*)(gp + i);
        sx += (xv.x + xv.y) + (xv.z + xv.w);
        sy += (xv.x * gv.x + xv.y * gv.y) + (xv.z * gv.z + xv.w * gv.w);
    }
    sdy[threadIdx.x] = sy;
    sdx[threadIdx.x] = sx;
    __syncthreads();
    for (int s = 128; s > 0; s >>= 1) {
        if (threadIdx.x < s) {
            sdy[threadIdx.x] += sdy[threadIdx.x + s];
            sdx[threadIdx.x] += sdx[threadIdx.x + s];
        }
        __syncthreads();
    }
    if (threadIdx.x == 0) {
        Sy[id] = sdy[0];
        Sx[id] = sdx[0];
    }
}

// ---------------------------------------------------------------------------
// Kernel 4: tiny attention conv (1,2,3,3) over (c',t) + sigmoid.
// yr channel 0 = shifted y-mean, channel 1 = r-mean; both derived from Sy/Sx.
// watt layout: (b, g, c', t)
// ---------------------------------------------------------------------------
__global__ void gsf_attn(const float* __restrict__ Sy,
                         const float* __restrict__ Sx,
                         const float* __restrict__ wc1,
                         const float* __restrict__ bc1,
                         const float* __restrict__ wc2,
                         const float* __restrict__ bc2,
                         float* __restrict__ watt)
{
    int idx = blockIdx.x * 256 + threadIdx.x;
    const int total = B_BATCH * 2 * C_HALF * T_SEG;
    if (idx >= total) return;

    const int t = idx & 7;
    const int c = (idx >> 3) & 127;
    const int g = (idx >> 10) & 1;
    const int b = idx >> 11;

    const float* wc = g ? wc2 : wc1;
    float acc = g ? bc2[0] : bc1[0];

    #pragma unroll
    for (int a = 0; a < 3; ++a) {
        int cc = c + a - 1;
        if (cc < 0 || cc >= C_HALF) continue;
        #pragma unroll
        for (int e = 0; e < 3; ++e) {
            int tt = t + e - 1;
            if (tt < 0 || tt >= T_SEG) continue;
            int sid = ((b * C_TOT + g * C_HALF + cc) << 3) + tt;
            float sy = Sy[sid];
            float sx = Sx[sid];
            float ym;
            if (g == 0) ym = (tt < T_SEG - 1) ? Sy[sid + 1] : 0.0f;  // y shifted -t
            else        ym = (tt > 0)         ? Sy[sid - 1] : 0.0f;  // y shifted +t
            ym *= INV_HW;
            float rm = (sx - sy) * INV_HW;
            acc += ym * wc[a * 3 + e] + rm * wc[9 + a * 3 + e];
        }
    }
    watt[idx] = 1.0f / (1.0f + expf(-acc));
}

// ---------------------------------------------------------------------------
// Kernel 5: final blend + channel interleave -> d_out.
// One thread per (b, c, 4-wide hw chunk); loops all 8 t with x/gate cached
// in registers so x is read exactly once despite the +-1 temporal shift.
// All global traffic is b128.
// ---------------------------------------------------------------------------
__global__ void gsf_final(const float* __restrict__ x,
                          const float* __restrict__ gate,
                          const float* __restrict__ watt,
                          float* __restrict__ out)
{
    long idx = (long)blockIdx.x * 256 + threadIdx.x;
    const long total = (long)B_BATCH * C_TOT * HW4;
    if (idx >= total) return;

    const int  hw = (int)(idx % HW4) * 4;
    long tmp = idx / HW4;
    const int  c  = (int)(tmp & 255);
    const int  b  = (int)(tmp >> 8);
    const int  g  = c >> 7;
    const int  cl = c & 127;
    const int  cout = g * C_HALF + ((cl & 63) << 1) + (cl >> 6);

    const long xtstr = (long)C_TOT * HW;          // t-stride in x / out
    const float* xp = x    + ((long)(b * T_SEG) * C_TOT + c) * (long)HW + hw;
    const float* gp = gate + ((long)(b * 2 + g) * T_SEG) * (long)HW + hw;
    float*       op = out  + ((long)(b * T_SEG) * C_TOT + cout) * (long)HW + hw;
    const float* wp = watt + ((long)((b * 2 + g) * C_HALF + cl) << 3);

    float4 xv[T_SEG], gv[T_SEG];
    #pragma unroll
    for (int t = 0; t < T_SEG; ++t) {
        xv[t] = *(const float4*)(xp + (long)t * xtstr);
        gv[t] = *(const float4*)(gp + (long)t * HW);
    }
    #pragma unroll
    for (int t = 0; t < T_SEG; ++t) {
        const float wt = wp[t];
        float4 rr, ys, o;
        rr.x = xv[t].x * (1.0f - gv[t].x);
        rr.y = xv[t].y * (1.0f - gv[t].y);
        rr.z = xv[t].z * (1.0f - gv[t].z);
        rr.w = xv[t].w * (1.0f - gv[t].w);
        if (g == 0) {
            if (t < T_SEG - 1) {
                ys.x = gv[t + 1].x * xv[t + 1].x;
                ys.y = gv[t + 1].y * xv[t + 1].y;
                ys.z = gv[t + 1].z * xv[t + 1].z;
                ys.w = gv[t + 1].w * xv[t + 1].w;
            } else ys = make_float4(0.f, 0.f, 0.f, 0.f);
        } else {
            if (t > 0) {
                ys.x = gv[t - 1].x * xv[t - 1].x;
                ys.y = gv[t - 1].y * xv[t - 1].y;
                ys.z = gv[t - 1].z * xv[t - 1].z;
                ys.w = gv[t - 1].w * xv[t - 1].w;
            } else ys = make_float4(0.f, 0.f, 0.f, 0.f);
        }
        o.x = rr.x + (ys.x - rr.x) * wt;
        o.y = rr.y + (ys.y - rr.y) * wt;
        o.z = rr.z + (ys.z - rr.z) * wt;
        o.w = rr.w + (ys.w - rr.w) * wt;
        *(float4*)(op + (long)t * xtstr) = o;
    }
}

// ---------------------------------------------------------------------------
// launch
// ---------------------------------------------------------------------------
static inline size_t align256(size_t v) { return (v + 255) & ~(size_t)255; }

extern "C" void kernel_launch(void* const* d_in, const int* in_sizes, int n_in,
                              void* d_out, int out_size, void* d_ws, size_t ws_size,
                              hipStream_t stream) {
    const float* x     = (const float*)d_in[0];
    const float* gamma = (const float*)d_in[1];
    const float* beta  = (const float*)d_in[2];
    const float* mean  = (const float*)d_in[3];
    const float* var   = (const float*)d_in[4];
    const float* w3d   = (const float*)d_in[5];
    const float* b3d   = (const float*)d_in[6];
    const float* wc1   = (const float*)d_in[7];
    const float* bc1   = (const float*)d_in[8];
    const float* wc2   = (const float*)d_in[9];
    const float* bc2   = (const float*)d_in[10];
    float* out = (float*)d_out;

    // workspace carve-up (~53 MB total)
    char* ws = (char*)d_ws;
    size_t off = 0;
    const size_t szP    = (size_t)B_BATCH * 2 * KPAD * T_SEG * HW * sizeof(float);
    const size_t szWt   = (size_t)2 * KPAD * C_HALF * sizeof(float);
    const size_t szSS   = (size_t)C_TOT * sizeof(float);
    const size_t szGate = (size_t)B_BATCH * 2 * T_SEG * HW * sizeof(float);
    const size_t szSum  = (size_t)B_BATCH * C_TOT * T_SEG * sizeof(float);
    const size_t szW    = (size_t)B_BATCH * 2 * C_HALF * T_SEG * sizeof(float);

    float* P     = (float*)(ws + off); off = align256(off + szP);
    float* Wt    = (float*)(ws + off); off = align256(off + szWt);
    float* scale = (float*)(ws + off); off = align256(off + szSS);
    float* shift = (float*)(ws + off); off = align256(off + szSS);
    float* gateb = (float*)(ws + off); off = align256(off + szGate);
    float* Syb   = (float*)(ws + off); off = align256(off + szSum);
    float* Sxb   = (float*)(ws + off); off = align256(off + szSum);
    float* watt  = (float*)(ws + off); off = align256(off + szW);

    // 0) BN coefficients + transposed tap-padded weights
    gsf_prep<<<32, 256, 0, stream>>>(gamma, beta, mean, var, w3d, Wt, scale, shift);

    // 1) tap-GEMM (WMMA f32 16x16x4): 16 (b,g) x 8 t x 56 h blocks, 4 waves each
    dim3 gGemm(H_DIM, T_SEG, B_BATCH * 2);
    gsf_gate_gemm<<<gGemm, 128, 0, stream>>>(x, Wt, scale, shift, P);

    // 2) 27-tap shifted reduce + tanh -> gate
    const int nGate = B_BATCH * 2 * T_SEG * HW;
    gsf_gate_reduce<<<(nGate + 255) / 256, 256, 0, stream>>>(P, b3d, gateb);

    // 3) spatial sums per (b,c,t)
    gsf_sums<<<B_BATCH * C_TOT * T_SEG, 256, 0, stream>>>(x, gateb, Syb, Sxb);

    // 4) attention weights
    const int nAtt = B_BATCH * 2 * C_HALF * T_SEG;
    gsf_attn<<<(nAtt + 255) / 256, 256, 0, stream>>>(Syb, Sxb, wc1, bc1, wc2, bc2, watt);

    // 5) final blend + interleave (all b128 traffic)
    const long nFin = (long)B_BATCH * C_TOT * HW4;
    gsf_final<<<(int)((nFin + 255) / 256), 256, 0, stream>>>(x, gateb, watt, out);
}